// GraphSAGEModel_42863773614470
// MI455X (gfx1250) — compile-verified
//
#include <hip/hip_runtime.h>

typedef float v2f __attribute__((ext_vector_type(2)));
typedef float v8f __attribute__((ext_vector_type(8)));

#define N_NODES_C 100000
#define N_EDGES_C 1600000
#define N_GRAPHS_C 64
#define DIM 128
#define N_CLS_C 8

// ---------------- zero (float4 grid-stride) ----------------
__global__ void zero_f4(float* __restrict__ p, long n4) {
  long i = (long)blockIdx.x * blockDim.x + threadIdx.x;
  long stride = (long)gridDim.x * blockDim.x;
  float4 z = make_float4(0.f, 0.f, 0.f, 0.f);
  for (; i < n4; i += stride) ((float4*)p)[i] = z;
}

// ---------------- edge scatter: agg[dst] += X[src], deg[dst] += 1 ----------------
// one wave32 per edge; lane l handles 4 consecutive floats of the 128-float row
__global__ void edge_scatter(const float* __restrict__ X,
                             const int* __restrict__ src,
                             const int* __restrict__ dst,
                             float* __restrict__ agg,
                             float* __restrict__ deg,
                             int n_edges, int add_deg) {
  int gtid = blockIdx.x * blockDim.x + threadIdx.x;
  int wid  = gtid >> 5;
  int lane = threadIdx.x & 31;
  int nw   = (gridDim.x * blockDim.x) >> 5;
  for (int e = wid; e < n_edges; e += nw) {
    int s = src[e];
    int d = dst[e];
    int e2 = e + nw;
    if (e2 < n_edges) {
      int s2 = src[e2];
      __builtin_prefetch(X + (size_t)s2 * DIM + lane * 4, 0, 0);  // global_prefetch_b8
    }
    float4 x = *(const float4*)(X + (size_t)s * DIM + lane * 4);
    float* a = agg + (size_t)d * DIM + lane * 4;
    atomicAdd(a + 0, x.x);
    atomicAdd(a + 1, x.y);
    atomicAdd(a + 2, x.z);
    atomicAdd(a + 3, x.w);
    if (add_deg && lane == 0) atomicAdd(deg + d, 1.0f);
  }
}

// ---------------- deg -> 1/max(deg,1) in place; graph counts ----------------
__global__ void rdeg_gcnt(float* __restrict__ deg, const int* __restrict__ gids,
                          float* __restrict__ gcnt, int n) {
  int i = blockIdx.x * blockDim.x + threadIdx.x;
  if (i < n) {
    deg[i] = 1.0f / fmaxf(deg[i], 1.0f);
    atomicAdd(gcnt + gids[i], 1.0f);
  }
}

// ---------------- fused SAGE layer GEMM with f32 WMMA ----------------
// out[16 nodes,128] = Xself@Wself + (Xneigh*rdeg)@Wneigh + bias, optional relu,
// either stored to `out` or atomic-accumulated into gsum[graph_id] (mean_nodes fusion).
__global__ __launch_bounds__(256)
void sage_gemm(const float* __restrict__ Xself,
               const float* __restrict__ Xneigh,
               const float* __restrict__ rdeg,
               const float* __restrict__ Wself,
               const float* __restrict__ Wneigh,
               const float* __restrict__ bias,
               float* __restrict__ out,
               float* __restrict__ gsum,
               const int* __restrict__ gids,
               int n_tiles, int do_relu, int to_gsum) {
  extern __shared__ float lds[];
  float* lWs = lds;            // 16384 floats, pair-swizzled: [(k>>1)*256 + n*2 + (k&1)]
  float* lWn = lds + 16384;    // 16384 floats
  float* lb  = lds + 32768;    // 128 floats

  // cooperative staging of both weight matrices (swizzled so a B fragment
  // {W[k][n], W[k+1][n]} is one contiguous 8-byte LDS read)
  for (int idx = threadIdx.x; idx < DIM * DIM; idx += blockDim.x) {
    int k = idx >> 7, n = idx & 127;
    int pos = (k >> 1) * 256 + n * 2 + (k & 1);
    lWs[pos] = Wself[idx];
    lWn[pos] = Wneigh[idx];
  }
  if (threadIdx.x < DIM) lb[threadIdx.x] = bias[threadIdx.x];
  __syncthreads();

  int lane = threadIdx.x & 31;
  int wib  = threadIdx.x >> 5;
  int m    = lane & 15;   // A row / C column within tile
  int hi   = lane >> 4;   // K-half for A/B, M-half for C/D
  int wstride = (gridDim.x * blockDim.x) >> 5;

  for (int t = blockIdx.x * (blockDim.x >> 5) + wib; t < n_tiles; t += wstride) {
    int r0  = t * 16;
    int row = r0 + m;
    float rd = rdeg[row];

    v8f acc[8];
#pragma unroll
    for (int nt = 0; nt < 8; ++nt) {
      float bv = lb[nt * 16 + m];
#pragma unroll
      for (int r = 0; r < 8; ++r) acc[nt][r] = bv;
    }

#pragma unroll 2
    for (int kk = 0; kk < DIM; kk += 4) {
      int kb = kk + 2 * hi;
      v2f a_s = *(const v2f*)(Xself  + (size_t)row * DIM + kb);
      v2f a_n = *(const v2f*)(Xneigh + (size_t)row * DIM + kb);
      a_n[0] *= rd;
      a_n[1] *= rd;
      int lbase = (kb >> 1) * 256 + m * 2;

      // batch ALL B fragments first -> one dscnt wait, then 16 straight WMMAs
      v2f bs[8], bn[8];
#pragma unroll
      for (int nt = 0; nt < 8; ++nt) {
        bs[nt] = *(const v2f*)(lWs + lbase + nt * 32);
        bn[nt] = *(const v2f*)(lWn + lbase + nt * 32);
      }
      // 8 independent accumulator chains (self), then the dependent neigh pass
#pragma unroll
      for (int nt = 0; nt < 8; ++nt) {
        acc[nt] = __builtin_amdgcn_wmma_f32_16x16x4_f32(
            false, a_s, false, bs[nt], (short)0, acc[nt], false, false);
      }
#pragma unroll
      for (int nt = 0; nt < 8; ++nt) {
        acc[nt] = __builtin_amdgcn_wmma_f32_16x16x4_f32(
            false, a_n, false, bn[nt], (short)0, acc[nt], false, false);
      }
    }

    if (!to_gsum) {
#pragma unroll
      for (int nt = 0; nt < 8; ++nt) {
#pragma unroll
        for (int r = 0; r < 8; ++r) {
          int rr = r0 + r + 8 * hi;
          float v = acc[nt][r];
          if (do_relu) v = fmaxf(v, 0.0f);
          out[(size_t)rr * DIM + nt * 16 + m] = v;
        }
      }
    } else {
      int gid_r[8];
#pragma unroll
      for (int r = 0; r < 8; ++r) gid_r[r] = gids[r0 + r + 8 * hi];
#pragma unroll
      for (int nt = 0; nt < 8; ++nt) {
#pragma unroll
        for (int r = 0; r < 8; ++r) {
          atomicAdd(gsum + gid_r[r] * DIM + nt * 16 + m, acc[nt][r]);
        }
      }
    }
  }
}

// ---------------- head: hg = gsum/cnt; out = [hg|perm] @ Wc + bc ----------------
__global__ void head_kernel(const float* __restrict__ gsum,
                            const float* __restrict__ gcnt,
                            const float* __restrict__ perm,
                            const float* __restrict__ Wc,
                            const float* __restrict__ bc,
                            float* __restrict__ out) {
  int t = threadIdx.x;       // 512 = 64 graphs * 8 classes
  int g = t >> 3;
  int c = t & 7;
  float inv = 1.0f / fmaxf(gcnt[g], 1.0f);
  float acc = bc[c];
  for (int k = 0; k < DIM; ++k) acc += gsum[g * DIM + k] * inv * Wc[k * N_CLS_C + c];
  for (int k = 0; k < DIM; ++k) acc += perm[g * DIM + k] * Wc[(DIM + k) * N_CLS_C + c];
  out[g * N_CLS_C + c] = acc;
}

extern "C" void kernel_launch(void* const* d_in, const int* in_sizes, int n_in,
                              void* d_out, int out_size, void* d_ws, size_t ws_size,
                              hipStream_t stream) {
  const float* h    = (const float*)d_in[0];
  const float* perm = (const float*)d_in[1];
  const int*   src  = (const int*)d_in[2];
  const int*   dst  = (const int*)d_in[3];
  const int*   gids = (const int*)d_in[4];
  const float* W1s  = (const float*)d_in[5];
  const float* W1n  = (const float*)d_in[6];
  const float* b1   = (const float*)d_in[7];
  const float* W2s  = (const float*)d_in[8];
  const float* W2n  = (const float*)d_in[9];
  const float* b2   = (const float*)d_in[10];
  const float* Wc   = (const float*)d_in[11];
  const float* bc   = (const float*)d_in[12];

  // workspace layout (floats): agg | gsum | gcnt | deg | h1   (~103 MB)
  float* ws   = (float*)d_ws;
  float* agg  = ws;
  float* gsum = agg  + (size_t)N_NODES_C * DIM;
  float* gcnt = gsum + (size_t)N_GRAPHS_C * DIM;
  float* deg  = gcnt + N_GRAPHS_C;
  float* h1   = deg  + N_NODES_C;

  size_t lds_bytes = (size_t)(2 * DIM * DIM + DIM) * sizeof(float);
  int n_tiles = N_NODES_C / 16;

  // zero agg + gsum + gcnt + deg (contiguous, multiple of 4 floats)
  long z1 = ((long)N_NODES_C * DIM + (long)N_GRAPHS_C * DIM + N_GRAPHS_C + N_NODES_C) / 4;
  zero_f4<<<2048, 256, 0, stream>>>(agg, z1);

  // layer 1: scatter, mean-scale prep, fused GEMM + relu -> h1
  edge_scatter<<<4096, 256, 0, stream>>>(h, src, dst, agg, deg, N_EDGES_C, 1);
  rdeg_gcnt<<<(N_NODES_C + 255) / 256, 256, 0, stream>>>(deg, gids, gcnt, N_NODES_C);
  sage_gemm<<<256, 256, lds_bytes, stream>>>(h, agg, deg, W1s, W1n, b1,
                                             h1, nullptr, nullptr, n_tiles, 1, 0);

  // layer 2: re-zero agg, scatter h1, fused GEMM straight into per-graph sums
  zero_f4<<<2048, 256, 0, stream>>>(agg, (long)N_NODES_C * DIM / 4);
  edge_scatter<<<4096, 256, 0, stream>>>(h1, src, dst, agg, deg, N_EDGES_C, 0);
  sage_gemm<<<256, 256, lds_bytes, stream>>>(h1, agg, deg, W2s, W2n, b2,
                                             nullptr, gsum, gids, n_tiles, 0, 1);

  // head
  head_kernel<<<1, 512, 0, stream>>>(gsum, gcnt, perm, Wc, bc, (float*)d_out);
}